// DynamicVoxelizer_8100308320786
// MI455X (gfx1250) — compile-verified
//
#include <hip/hip_runtime.h>
#include <math.h>

// ---------------------------------------------------------------------------
// DynamicVoxelizer for MI455X (gfx1250, wave32).
// Memory-bound streaming kernel: 12 B in / 44 B out per point (~448 MB total
// -> ~19 us at 23.3 TB/s HBM; this kernel is pure bandwidth).
// Uses gfx1250 async-to-LDS copies (ASYNCcnt) for the input stream and
// non-temporal 128-bit stores for all outputs.
// ---------------------------------------------------------------------------

typedef float v4f __attribute__((ext_vector_type(4)));
typedef int v4i __attribute__((ext_vector_type(4)));

#define TPB 256
#define GROUPS_PER_THREAD 4   // each group = 4 points = 3 x float4

#if defined(__has_builtin)
#if __has_builtin(__builtin_amdgcn_global_load_async_to_lds_b128)
#define HAS_ASYNC_LDS 1
#endif
#endif
#ifndef HAS_ASYNC_LDS
#define HAS_ASYNC_LDS 0
#endif

#if HAS_ASYNC_LDS
typedef __attribute__((address_space(1))) v4i g_v4i;  // global int4*
typedef __attribute__((address_space(3))) v4i l_v4i;  // LDS int4*
#if __has_builtin(__builtin_amdgcn_s_wait_asynccnt)
#define WAIT_ASYNC(n) __builtin_amdgcn_s_wait_asynccnt(n)
#else
#define WAIT_ASYNC(n) asm volatile("s_wait_asynccnt %0" ::"i"(n) : "memory")
#endif
#define WAIT_DS0() asm volatile("s_wait_dscnt 0x0" ::: "memory")
#define LDS_FENCE() asm volatile("" ::: "memory")
#endif

__global__ __launch_bounds__(TPB) void voxelize_kernel(
    const float* __restrict__ pts, float* __restrict__ out,
    int ngroups, int P) {
  // Output layout (floats): points[3P] coords[3P] idx[P] offsets[3P] valid[P]
  float* __restrict__ oP = out;
  float* __restrict__ oC = out + 3L * (long)P;
  float* __restrict__ oI = out + 6L * (long)P;
  float* __restrict__ oO = out + 7L * (long)P;
  float* __restrict__ oV = out + 10L * (long)P;

  const v4f* __restrict__ pv = (const v4f*)pts;
  const long nvec = 3L * (long)ngroups;  // total float4's of input

  const int tid = blockIdx.x * TPB + threadIdx.x;
  const int stride = gridDim.x * TPB;

#if HAS_ASYNC_LDS
  // Per wave: 2 buffers x 96 float4 (32 lanes x 3 chunks) = 3 KB; 24 KB/block.
  __shared__ v4f lds[(TPB / 32) * 2 * 96];
  const int lane = threadIdx.x & 31;
  v4f* wreg = &lds[(threadIdx.x >> 5) * 192];
#endif

  auto st_v4 = [&](float* basep, float f0, float f1, float f2, float f3) {
    v4f v = {f0, f1, f2, f3};
    __builtin_nontemporal_store(v, (v4f*)basep);
  };

  auto process = [&](int g, v4f a, v4f b, v4f c) {
    float pt[12] = {a[0], a[1], a[2], a[3], b[0], b[1], b[2], b[3],
                    c[0], c[1], c[2], c[3]};
    float op[12], oc[12], oo[12], oi[4], ov[4];
    const int n0 = (g * 4) % 1000000;  // index within batch (N = 1e6, 4 | N)
#pragma unroll
    for (int j = 0; j < 4; ++j) {
      float x = pt[3 * j + 0];
      float y = pt[3 * j + 1];
      float z = pt[3 * j + 2];
      bool nn = !((x != x) | (y != y) | (z != z));
      float px = nn ? x : 0.0f;
      float py = nn ? y : 0.0f;
      float pz = nn ? z : 0.0f;
      int cx = (int)floorf((px + 51.2f) / 0.2f);
      int cy = (int)floorf((py + 51.2f) / 0.2f);
      int cz = (int)floorf((pz + 3.0f) / 0.2f);
      bool inr = (cx >= 0) & (cx < 512) & (cy >= 0) & (cy < 512) & (cz >= 0) &
                 (cz < 30);
      bool v = nn & inr;
      op[3 * j + 0] = v ? px : 0.0f;
      op[3 * j + 1] = v ? py : 0.0f;
      op[3 * j + 2] = v ? pz : 0.0f;
      oc[3 * j + 0] = v ? (float)cz : -1.0f;   // zyx order
      oc[3 * j + 1] = v ? (float)cy : -1.0f;
      oc[3 * j + 2] = v ? (float)cx : -1.0f;
      float ctx = (float)cx * 0.2f - 51.2f + 0.1f;
      float cty = (float)cy * 0.2f - 51.2f + 0.1f;
      float ctz = (float)cz * 0.2f - 3.0f + 0.1f;
      oo[3 * j + 0] = v ? (px - ctx) : 0.0f;
      oo[3 * j + 1] = v ? (py - cty) : 0.0f;
      oo[3 * j + 2] = v ? (pz - ctz) : 0.0f;
      oi[j] = v ? (float)(n0 + j) : -1.0f;
      ov[j] = v ? 1.0f : 0.0f;
    }
    const long f12 = 12L * (long)g;  // float offset of this group's 12-float block
    const long f4 = 4L * (long)g;
#pragma unroll
    for (int cch = 0; cch < 3; ++cch) {
      st_v4(oP + f12 + 4 * cch, op[4 * cch], op[4 * cch + 1], op[4 * cch + 2],
            op[4 * cch + 3]);
      st_v4(oC + f12 + 4 * cch, oc[4 * cch], oc[4 * cch + 1], oc[4 * cch + 2],
            oc[4 * cch + 3]);
      st_v4(oO + f12 + 4 * cch, oo[4 * cch], oo[4 * cch + 1], oo[4 * cch + 2],
            oo[4 * cch + 3]);
    }
    st_v4(oI + f4, oi[0], oi[1], oi[2], oi[3]);
    st_v4(oV + f4, ov[0], ov[1], ov[2], ov[3]);
  };

#if HAS_ASYNC_LDS
  // Wave-contiguous async staging: instruction c copies float4 indices
  // [3*wave_start + 32c, +32) -> one fully contiguous 512B burst per wave.
  auto prefetch = [&](int g, int buf) {
    long base = 3L * (long)(g - lane);  // wave_start * 3 (uniform per wave)
    WAIT_DS0();  // close DS-read vs async-LDS-write WAR hazard on this buffer
#pragma unroll
    for (int c = 0; c < 3; ++c) {
      long idx = base + c * 32 + lane;
      idx = idx < (nvec - 1) ? idx : (nvec - 1);  // clamp tail reads in-bounds
      const v4f* src = pv + idx;
      v4f* dst = wreg + buf * 96 + c * 32 + lane;
      __builtin_amdgcn_global_load_async_to_lds_b128((g_v4i*)src, (l_v4i*)dst,
                                                     0, 0);
    }
  };

  int g = tid;
  int buf = 0;
  if (g < ngroups) prefetch(g, 0);
  for (; g < ngroups; g += stride, buf ^= 1) {
    int gn = g + stride;
    if (gn < ngroups) {
      prefetch(gn, buf ^ 1);
      WAIT_ASYNC(3);  // previous buffer's 3 copies complete
    } else {
      WAIT_ASYNC(0);
    }
    LDS_FENCE();
    v4f a = wreg[buf * 96 + 3 * lane + 0];
    v4f b = wreg[buf * 96 + 3 * lane + 1];
    v4f c = wreg[buf * 96 + 3 * lane + 2];
    process(g, a, b, c);
  }
#else
  for (int g = tid; g < ngroups; g += stride) {
    const v4f* src = pv + 3L * (long)g;
    v4f a = __builtin_nontemporal_load(src + 0);
    v4f b = __builtin_nontemporal_load(src + 1);
    v4f c = __builtin_nontemporal_load(src + 2);
    process(g, a, b, c);
  }
#endif
}

extern "C" void kernel_launch(void* const* d_in, const int* in_sizes, int n_in,
                              void* d_out, int out_size, void* d_ws,
                              size_t ws_size, hipStream_t stream) {
  (void)n_in; (void)out_size; (void)d_ws; (void)ws_size;
  const float* pts = (const float*)d_in[0];
  float* out = (float*)d_out;
  long total_f = (long)in_sizes[0];   // 8 * 1e6 * 3
  int P = (int)(total_f / 3);         // total points (8M)
  int ngroups = P / 4;                // P is a multiple of 4
  int denom = TPB * GROUPS_PER_THREAD;
  int blocks = (ngroups + denom - 1) / denom;  // 1954 for the reference shape
  if (blocks < 1) blocks = 1;
  voxelize_kernel<<<blocks, TPB, 0, stream>>>(pts, out, ngroups, P);
}